// Coloring_GraphSAGE_32779190403249
// MI455X (gfx1250) — compile-verified
//
#include <hip/hip_runtime.h>

// ---------------------------------------------------------------------------
// GraphSAGE (6x SAGEConv, mean aggregation) for MI455X / gfx1250.
//   - edge scatter: native fp32 L2 atomics (agg buffer is L2-resident: 12.8MB)
//   - dense layers: V_WMMA_F32_16X16X4_F32, one 16x16 output tile per wave32
//   - 1/deg scale, bias, ReLU fused into the WMMA producer/epilogue
// ---------------------------------------------------------------------------

typedef float v2f __attribute__((ext_vector_type(2)));
typedef float v8f __attribute__((ext_vector_type(8)));

#define NNODES 50000
#define NEDGES 800000

__global__ void fill_zero_kernel(float* __restrict__ p, int n) {
  int i = blockIdx.x * blockDim.x + threadIdx.x;
  if (i < n) p[i] = 0.0f;
}

__global__ void deg_kernel(const int* __restrict__ dst, float* __restrict__ deg, int E) {
  int e = blockIdx.x * blockDim.x + threadIdx.x;
  if (e < E) {
    __hip_atomic_fetch_add(&deg[dst[e]], 1.0f, __ATOMIC_RELAXED, __HIP_MEMORY_SCOPE_AGENT);
  }
}

// One thread per (edge, 4-channel quad): float4 gather + 4 fp32 atomic adds.
template <int D>
__global__ void scatter_add_kernel(const float* __restrict__ x, const int* __restrict__ src,
                                   const int* __restrict__ dst, float* __restrict__ agg, int E) {
  constexpr int QUADS = D / 4;                 // 8 or 16
  constexpr int QSHIFT = (D == 32) ? 3 : 4;
  long long tid = (long long)blockIdx.x * blockDim.x + threadIdx.x;
  if (tid >= (long long)E * QUADS) return;
  int e = (int)(tid >> QSHIFT);
  int q = (int)(tid & (QUADS - 1));
  int s = src[e];
  int d = dst[e];
  const float4 v = *(const float4*)(x + (long long)s * D + q * 4);
  float* a = agg + (long long)d * D + q * 4;
  __hip_atomic_fetch_add(a + 0, v.x, __ATOMIC_RELAXED, __HIP_MEMORY_SCOPE_AGENT);
  __hip_atomic_fetch_add(a + 1, v.y, __ATOMIC_RELAXED, __HIP_MEMORY_SCOPE_AGENT);
  __hip_atomic_fetch_add(a + 2, v.z, __ATOMIC_RELAXED, __HIP_MEMORY_SCOPE_AGENT);
  __hip_atomic_fetch_add(a + 3, v.w, __ATOMIC_RELAXED, __HIP_MEMORY_SCOPE_AGENT);
}

// out[n, :] = relu( (agg[n,:]/max(deg,1)) @ Wl^T + bl + xin[n,:] @ Wr^T )
// One wave32 per 16x16 output tile; fp32 WMMA 16x16x4, K fully unrolled.
template <int DIN, int DOUT>
__global__ void sage_gemm_kernel(const float* __restrict__ agg, const float* __restrict__ deg,
                                 const float* __restrict__ xin,
                                 const float* __restrict__ Wl, const float* __restrict__ bl,
                                 const float* __restrict__ Wr,
                                 float* __restrict__ out, int nWaves) {
  const int wave = blockIdx.x * (blockDim.x >> 5) + (threadIdx.x >> 5);
  if (wave >= nWaves) return;                  // wave-uniform: EXEC stays all-1s for WMMA
  const int lane = threadIdx.x & 31;
  constexpr int COL_TILES = DOUT / 16;
  const int rowTile = wave / COL_TILES;
  const int colTile = wave - rowTile * COL_TILES;
  const int row0 = rowTile * 16;
  const int col0 = colTile * 16;

  // A fragment: row m = lane&15, K pair base kh = (lane>>4)*2  (ISA 7.12.2, f32 16x4 A)
  // B fragment: col n = lane&15, same K pair                    (f32 4x16 B)
  const int m  = lane & 15;
  const int kh = (lane >> 4) * 2;
  const int node = row0 + m;
  const int ncol = col0 + m;

  const float ds = 1.0f / fmaxf(deg[node], 1.0f);
  const float* __restrict__ aMean = agg + (long long)node * DIN;
  const float* __restrict__ aX    = xin + (long long)node * DIN;
  const float* __restrict__ wlRow = Wl + (long long)ncol * DIN;   // B = Wl^T tile
  const float* __restrict__ wrRow = Wr + (long long)ncol * DIN;   // B = Wr^T tile

  v8f acc = {};
#pragma unroll
  for (int k = 0; k < DIN; k += 4) {           // acc += mean x Wl^T
    v2f a, b;
    a.x = aMean[k + kh] * ds;
    a.y = aMean[k + kh + 1] * ds;
    b.x = wlRow[k + kh];
    b.y = wlRow[k + kh + 1];
    acc = __builtin_amdgcn_wmma_f32_16x16x4_f32(false, a, false, b, (short)0, acc,
                                                false, false);
  }
#pragma unroll
  for (int k = 0; k < DIN; k += 4) {           // acc += x x Wr^T
    v2f a, b;
    a.x = aX[k + kh];
    a.y = aX[k + kh + 1];
    b.x = wrRow[k + kh];
    b.y = wrRow[k + kh + 1];
    acc = __builtin_amdgcn_wmma_f32_16x16x4_f32(false, a, false, b, (short)0, acc,
                                                false, false);
  }

  // D layout: lane holds col n = lane&15, rows (lane>>4)*8 + r   (ISA 7.12.2, f32 16x16 C/D)
  const float bias = bl[ncol];
  const int n = col0 + (lane & 15);
  const int rbase = (lane >> 4) * 8;
#pragma unroll
  for (int r = 0; r < 8; ++r) {
    const int row = row0 + rbase + r;
    const float v = acc[r] + bias;
    out[(long long)row * DOUT + n] = fmaxf(v, 0.0f);
  }
}

// softmax over first 5 cols of h (stride 64); write to out5 and patch h in place
__global__ void softmax_head5_kernel(float* __restrict__ h, float* __restrict__ out5, int N) {
  int nidx = blockIdx.x * blockDim.x + threadIdx.x;
  if (nidx >= N) return;
  float v[5];
  float mx = -3.0e38f;
#pragma unroll
  for (int j = 0; j < 5; ++j) { v[j] = h[(long long)nidx * 64 + j]; mx = fmaxf(mx, v[j]); }
  float s = 0.0f;
#pragma unroll
  for (int j = 0; j < 5; ++j) { v[j] = __expf(v[j] - mx); s += v[j]; }
  const float inv = 1.0f / s;
#pragma unroll
  for (int j = 0; j < 5; ++j) {
    const float r = v[j] * inv;
    out5[(long long)nidx * 5 + j] = r;
    h[(long long)nidx * 64 + j] = r;
  }
}

// full softmax over 32 cols (stride 32)
__global__ void softmax32_kernel(const float* __restrict__ h, float* __restrict__ out, int N) {
  int nidx = blockIdx.x * blockDim.x + threadIdx.x;
  if (nidx >= N) return;
  float v[32];
  float mx = -3.0e38f;
#pragma unroll
  for (int j = 0; j < 32; ++j) { v[j] = h[(long long)nidx * 32 + j]; mx = fmaxf(mx, v[j]); }
  float s = 0.0f;
#pragma unroll
  for (int j = 0; j < 32; ++j) { v[j] = __expf(v[j] - mx); s += v[j]; }
  const float inv = 1.0f / s;
#pragma unroll
  for (int j = 0; j < 32; ++j) out[(long long)nidx * 32 + j] = v[j] * inv;
}

extern "C" void kernel_launch(void* const* d_in, const int* in_sizes, int n_in,
                              void* d_out, int out_size, void* d_ws, size_t ws_size,
                              hipStream_t stream) {
  (void)in_sizes; (void)out_size; (void)ws_size;
  const int N = NNODES, E = NEDGES;

  const float* x = (const float*)d_in[0];
  const float* Wl[6]; const float* bl[6]; const float* Wr[6];
  for (int i = 0; i < 6; ++i) {
    Wl[i] = (const float*)d_in[1 + 3 * i];
    bl[i] = (const float*)d_in[2 + 3 * i];
    Wr[i] = (const float*)d_in[3 + 3 * i];
  }
  const int* ei  = (const int*)d_in[n_in - 1];   // edge_index [2, E] (int32)
  const int* src = ei;
  const int* dst = ei + E;
  float* out = (float*)d_out;

  // workspace layout (fp32): deg[N] | agg[N*64] | h0[N*64] | h1[N*64]  (~38.6 MB)
  float* ws  = (float*)d_ws;
  float* deg = ws;
  float* agg = ws + 50176;                       // N rounded up, 256B-aligned
  float* h0  = agg + (long long)N * 64;
  float* h1  = h0 + (long long)N * 64;

  const int TB = 256;

  // degree (constant across layers)
  fill_zero_kernel<<<(N + TB - 1) / TB, TB, 0, stream>>>(deg, N);
  deg_kernel<<<(E + TB - 1) / TB, TB, 0, stream>>>(dst, deg, E);

  const float* cur = x;
  float* nxt = h0;
  const float* last = nullptr;

  for (int L = 0; L < 6; ++L) {
    const int din  = (L == 0) ? 32 : 64;
    const int dout = (L == 5) ? 32 : 64;

    const int aggN = N * din;
    fill_zero_kernel<<<(aggN + TB - 1) / TB, TB, 0, stream>>>(agg, aggN);

    const long long scatThreads = (long long)E * (din / 4);
    const int scatBlocks = (int)((scatThreads + TB - 1) / TB);
    if (din == 32)
      scatter_add_kernel<32><<<scatBlocks, TB, 0, stream>>>(cur, src, dst, agg, E);
    else
      scatter_add_kernel<64><<<scatBlocks, TB, 0, stream>>>(cur, src, dst, agg, E);

    const int nWaves = (N / 16) * (dout / 16);
    const int gemmBlocks = (nWaves * 32 + TB - 1) / TB;
    if (din == 32 && dout == 64)
      sage_gemm_kernel<32, 64><<<gemmBlocks, TB, 0, stream>>>(agg, deg, cur, Wl[L], bl[L], Wr[L], nxt, nWaves);
    else if (din == 64 && dout == 64)
      sage_gemm_kernel<64, 64><<<gemmBlocks, TB, 0, stream>>>(agg, deg, cur, Wl[L], bl[L], Wr[L], nxt, nWaves);
    else
      sage_gemm_kernel<64, 32><<<gemmBlocks, TB, 0, stream>>>(agg, deg, cur, Wl[L], bl[L], Wr[L], nxt, nWaves);

    if (L == 2)  // x_5 = softmax(x[:, :5]) -> first output region + patch hidden state
      softmax_head5_kernel<<<(N + TB - 1) / TB, TB, 0, stream>>>(nxt, out, N);

    last = nxt;
    cur = nxt;
    nxt = (nxt == h0) ? h1 : h0;
  }

  // final: softmax over 32 dims -> second output region
  softmax32_kernel<<<(N + TB - 1) / TB, TB, 0, stream>>>(last, out + (long long)N * 5, N);
}